// CapsLayer_51075751084570
// MI455X (gfx1250) — compile-verified
//
#include <hip/hip_runtime.h>

// ---------------------------------------------------------------------------
// Capsule routing on gfx1250.
//   inputs: [B=32][N=2048][K=16] f32      W: [N=2048][J=32][D=32][K=16] f32
//   uji[b,n,j,d] = sum_k W[n,j,d,k] * inputs[b,n,k]
// Strategy: never materialize uji (268MB). Recompute it per routing pass with
// V_WMMA_F32_16X16X4_F32; W (134MB) is L2-resident after the first pass.
// Logits are cumulative & linear in vj, so pass2 uses dot(u, vj0+vj1) and no
// 8MB logit tensor is needed. Only scratch: s accumulator + vj snapshots.
// ---------------------------------------------------------------------------

typedef float v2f __attribute__((ext_vector_type(2)));
typedef float v8f __attribute__((ext_vector_type(8)));

#define N_IN    2048
#define D_IN    16
#define NJ      32
#define ND      32
#define M_TOT   (NJ * ND)          // 1024 rows (j*32+d) of the per-n GEMM
#define PITCH   17                 // LDS row pitch (bank-conflict padding)
#define BLOCK   512                // 16 wave32 waves
#define NCHUNK  128                // n-chunks -> grid.x; chunk = 16 n's
#define CHUNK   (N_IN / NCHUNK)
#define S_ELEMS 32768              // 32*32*32 floats
#define SMEM_FLOATS (M_TOT * PITCH + 16 * 33)

// PHASE 0: cj uniform (1/32)  -> s0
// PHASE 1: logits = u . vj0   -> s1
// PHASE 2: logits = u . (vj0+vj1) -> s2
template <int PHASE>
__global__ __launch_bounds__(BLOCK, 1)
void caps_pass(const float* __restrict__ inp,
               const float* __restrict__ Wg,
               const float* __restrict__ vj0,
               const float* __restrict__ vj1,
               float* __restrict__ s_buf)
{
  extern __shared__ float smem[];
  float* lds_u  = smem;                  // [1024][PITCH] : u[jd][b_local]
  float* lds_sm = smem + M_TOT * PITCH;  // [16][33]      : softmax row sums

  const int tid   = threadIdx.x;
  const int wave  = tid >> 5;            // 0..15
  const int lane  = tid & 31;
  const int lhalf = lane & 15;
  const int hi    = lane >> 4;           // lane half selects K pair (ISA 16x4 layout)
  const int bloc  = tid & 15;            // owned batch column (routing phase)
  const int jown  = tid >> 4;            // owned capsule j (0..31)
  const int bh    = blockIdx.y;          // batch half (0/1)
  const int bglob = bh * 16;

  // vj for my (b, j) lives in registers; pass2 pre-sums vj0+vj1 (linearity).
  float vjreg[ND];
  if (PHASE >= 1) {
    const float* vp = vj0 + ((size_t)(bglob + bloc) * NJ + jown) * ND;
#pragma unroll
    for (int d = 0; d < ND; ++d) vjreg[d] = vp[d];
    if (PHASE == 2) {
      const float* vq = vj1 + ((size_t)(bglob + bloc) * NJ + jown) * ND;
#pragma unroll
      for (int d = 0; d < ND; ++d) vjreg[d] += vq[d];
    }
  }

  float sacc[ND];
#pragma unroll
  for (int d = 0; d < ND; ++d) sacc[d] = 0.0f;

  const int nbase = blockIdx.x * CHUNK;

  for (int nn = 0; nn < CHUNK; ++nn) {
    const int n = nbase + nn;
    const float* wn = Wg + (size_t)n * (M_TOT * D_IN);
    const float* xn = inp + ((size_t)(bglob + lhalf) * N_IN + n) * D_IN;

    // Speculative prefetch of next n's 64KB W slice (128B per thread).
    __builtin_prefetch(wn + M_TOT * D_IN + tid * 32, 0, 0);

    // ---- GEMM: 4 m-tiles per wave, K=16 via 4x V_WMMA_F32_16X16X4_F32 ----
    v8f acc[4];
#pragma unroll
    for (int i = 0; i < 4; ++i) {
#pragma unroll
      for (int r = 0; r < 8; ++r) acc[i][r] = 0.0f;
    }

#pragma unroll
    for (int s = 0; s < 4; ++s) {
      // ISA 32-bit 16x4 A layout: lanes 0-15 hold K={4s,4s+1}, lanes 16-31
      // hold K={4s+2,4s+3}; B (4x16) assumed symmetric.
      const int kk = 4 * s + 2 * hi;
      const v2f bf = *(const v2f*)(xn + kk);  // X[k][b]: b = lhalf
#pragma unroll
      for (int i = 0; i < 4; ++i) {
        const int m = (wave * 4 + i) * 16 + lhalf;  // jd row
        const v2f af = *(const v2f*)(wn + m * D_IN + kk);
        acc[i] = __builtin_amdgcn_wmma_f32_16x16x4_f32(
            false, af, false, bf, (short)0, acc[i], false, false);
      }
    }

    // Stage C tiles to LDS as u[jd][b]. C layout: VGPR r -> (M = r or 8+r,
    // N = lane&15) per lane half.
#pragma unroll
    for (int i = 0; i < 4; ++i) {
      const int mb = (wave * 4 + i) * 16 + hi * 8;
#pragma unroll
      for (int r = 0; r < 8; ++r)
        lds_u[(mb + r) * PITCH + lhalf] = acc[i][r];
    }
    __syncthreads();

    // ---- routing math: this thread owns (b=bloc, j=jown), all d ----
    float u[ND];
#pragma unroll
    for (int d = 0; d < ND; ++d)
      u[d] = lds_u[(jown * ND + d) * PITCH + bloc];  // conflict-free (pitch 17)

    float c;
    if (PHASE == 0) {
      c = 1.0f;  // softmax of zeros over J; 1/J folded into final scale
    } else {
      float lg = 0.0f;
#pragma unroll
      for (int d = 0; d < ND; ++d) lg = fmaf(u[d], vjreg[d], lg);
      const float e = __expf(lg);  // logits are O(few): no max-shift needed
      lds_sm[bloc * 33 + jown] = e;
      __syncthreads();
      float ssum = 0.0f;
#pragma unroll
      for (int jj = 0; jj < NJ; ++jj) ssum += lds_sm[bloc * 33 + jj];
      c = e / ssum;
    }

#pragma unroll
    for (int d = 0; d < ND; ++d) sacc[d] = fmaf(c, u[d], sacc[d]);

    __syncthreads();  // guard lds_u / lds_sm reuse next iteration
  }

  // Cross-workgroup reduction over n-chunks into the 128KB (L2-resident) s.
  const float scale = (PHASE == 0) ? (1.0f / 32.0f) : 1.0f;
  float* sp = s_buf + ((size_t)(bglob + bloc) * NJ + jown) * ND;
#pragma unroll
  for (int d = 0; d < ND; ++d)
    atomicAdd(sp + d, sacc[d] * scale);  // global_atomic_add_f32
}

__global__ void zero_k(float* __restrict__ p, int n) {
  const int i = blockIdx.x * blockDim.x + threadIdx.x;
  if (i < n) p[i] = 0.0f;
}

// Elementwise squash (reference sums squares over a singleton axis):
// v = (s^2/(1+s^2)) * s / sqrt(s^2 + 1e-7). Optionally re-zeroes s for the
// next pass (deterministic within one launch sequence).
__global__ void squash_k(const float* __restrict__ s_in,
                         float* __restrict__ v_out,
                         float* __restrict__ zero_buf, int n) {
  const int i = blockIdx.x * blockDim.x + threadIdx.x;
  if (i < n) {
    const float s  = s_in[i];
    const float s2 = s * s;
    v_out[i] = (s2 / (1.0f + s2)) * s * rsqrtf(s2 + 1e-7f);
    if (zero_buf) zero_buf[i] = 0.0f;
  }
}

extern "C" void kernel_launch(void* const* d_in, const int* in_sizes, int n_in,
                              void* d_out, int out_size, void* d_ws, size_t ws_size,
                              hipStream_t stream) {
  const float* inp = (const float*)d_in[0];   // [32][2048][16]
  const float* Wg  = (const float*)d_in[1];   // [1][2048][32][32][16]
  float* out   = (float*)d_out;               // [32][1][32][32]
  float* s_buf = (float*)d_ws;                // 128KB accumulator
  float* vj0   = s_buf + S_ELEMS;
  float* vj1   = vj0 + S_ELEMS;

  const dim3 grid(NCHUNK, 2);
  const size_t shmem = SMEM_FLOATS * sizeof(float);
  const int eb = 256, eg = (S_ELEMS + eb - 1) / eb;

  zero_k<<<eg, eb, 0, stream>>>(s_buf, S_ELEMS);
  caps_pass<0><<<grid, BLOCK, shmem, stream>>>(inp, Wg, nullptr, nullptr, s_buf);
  squash_k<<<eg, eb, 0, stream>>>(s_buf, vj0, s_buf, S_ELEMS);
  caps_pass<1><<<grid, BLOCK, shmem, stream>>>(inp, Wg, vj0, nullptr, s_buf);
  squash_k<<<eg, eb, 0, stream>>>(s_buf, vj1, s_buf, S_ELEMS);
  caps_pass<2><<<grid, BLOCK, shmem, stream>>>(inp, Wg, vj0, vj1, s_buf);
  squash_k<<<eg, eb, 0, stream>>>(s_buf, out, nullptr, S_ELEMS);
}